// VQVAE_88682484728326
// MI455X (gfx1250) — compile-verified
//
#include <hip/hip_runtime.h>

#define DIMC  64
#define BOOK  1024
#define EMB   64
#define BATCH 512
#define DECAYF 0.999f
#define GAINF  (1.0f - 0.999f)
#define EPSF   1e-6f

typedef __attribute__((ext_vector_type(2))) float v2f;
typedef __attribute__((ext_vector_type(4))) float f4;
typedef __attribute__((ext_vector_type(8))) float v8f;

// ---------------- zero workspace (counts + update_dict) ----------------
__global__ void vq_zero_kernel(float* __restrict__ p, int n) {
    int i = (blockIdx.x * blockDim.x + threadIdx.x) * 4;
    if (i < n) {
        f4 z = {0.f, 0.f, 0.f, 0.f};
        *(f4*)(p + i) = z;
    }
}

// ---------------- b_sq[d][k] = sum_e codebook[d][k][e]^2 ----------------
__global__ void vq_bsq_kernel(const float* __restrict__ cb, float* __restrict__ bsq) {
    int dk = blockIdx.x * 256 + threadIdx.x;          // 65536 rows
    const f4* row = (const f4*)(cb + (size_t)dk * EMB);
    float s = 0.f;
#pragma unroll
    for (int j = 0; j < 16; ++j) {
        f4 v = row[j];
        s += v.x * v.x + v.y * v.y + v.z * v.z + v.w * v.w;
    }
    bsq[dk] = s;
}

// ---------------- main: WMMA distances + argmin + fused outputs ----------------
// grid = (BATCH/16, DIMC), block = 256 (8 waves). Block owns (d, 16 batch rows).
// Wave w covers codes [w*128, (w+1)*128).
__launch_bounds__(256)
__global__ void vq_main_kernel(const float* __restrict__ x,
                               const float* __restrict__ cb,
                               const float* __restrict__ bsq,
                               float* __restrict__ cw,
                               float* __restrict__ onehot,
                               float* __restrict__ counts,
                               float* __restrict__ upd) {
    __shared__ float lx[16 * 66];      // A tile, padded stride 66 -> bank-conflict-free
    __shared__ float wmin[8][16];
    __shared__ int   widx[8][16];
    __shared__ int   fidx[16];

    const int b0   = blockIdx.x * 16;
    const int d    = blockIdx.y;
    const int t    = threadIdx.x;
    const int wave = t >> 5;
    const int lane = t & 31;
    const int half = lane >> 4;        // 0: K0/K1 lanes, 1: K2/K3 lanes (f32 A layout)
    const int mrow = lane & 15;

    // ---- stage 16x64 x-tile into LDS (each thread: one float4) ----
    {
        int m = t >> 4;
        int e = (t & 15) * 4;
        f4 v = *(const f4*)(x + (size_t)(b0 + m) * (DIMC * EMB) + d * EMB + e);
        float* q = &lx[m * 66 + e];
        q[0] = v.x; q[1] = v.y; q[2] = v.z; q[3] = v.w;
    }
    __syncthreads();

    const float* cbd  = cb  + (size_t)d * BOOK * EMB;
    const float* bsqd = bsq + d * BOOK;

    // ---- A fragments: 16 x (16x4 f32), e-block kk covers e = 4kk..4kk+3 ----
    v2f afrag[16];
#pragma unroll
    for (int kk = 0; kk < 16; ++kk) {
        int e = 4 * kk + 2 * half;
        v2f a;
        a.x = lx[mrow * 66 + e];
        a.y = lx[mrow * 66 + e + 1];
        afrag[kk] = a;
    }

    float runmin[8];
    int   runidx[8];
#pragma unroll
    for (int r = 0; r < 8; ++r) { runmin[r] = 3.4e38f; runidx[r] = 0; }

    // ---- loop over this wave's 8 code tiles of 16 ----
#pragma unroll 2
    for (int tile = 0; tile < 8; ++tile) {
        int n0   = wave * 128 + tile * 16;
        int code = n0 + mrow;
        const float* brow = cbd + (size_t)code * EMB + 2 * half;

        // Prefetch next tile's codebook row for this lane (2 cachelines / 256B row).
        if (tile < 7) {
            __builtin_prefetch(brow + 16 * EMB, 0, 3);
            __builtin_prefetch(brow + 16 * EMB + 32, 0, 3);
        }

        // Issue all 16 B-fragment loads as one clause (16 loads in flight, one wait).
        v2f bfrag[16];
#pragma unroll
        for (int kk = 0; kk < 16; ++kk)
            bfrag[kk] = *(const v2f*)(brow + 4 * kk);   // codebook[d][code][4kk+2h .. +1]

        float bq = bsqd[code];

        // Two independent accumulation chains halve the serial WMMA dependency depth.
        v8f c0 = {0.f, 0.f, 0.f, 0.f, 0.f, 0.f, 0.f, 0.f};
        v8f c1 = {0.f, 0.f, 0.f, 0.f, 0.f, 0.f, 0.f, 0.f};
#pragma unroll
        for (int kk = 0; kk < 16; kk += 2) {
            c0 = __builtin_amdgcn_wmma_f32_16x16x4_f32(
                    false, afrag[kk],     false, bfrag[kk],     (short)0, c0, false, false);
            c1 = __builtin_amdgcn_wmma_f32_16x16x4_f32(
                    false, afrag[kk + 1], false, bfrag[kk + 1], (short)0, c1, false, false);
        }

#pragma unroll
        for (int r = 0; r < 8; ++r) {
            float v = bq - 2.0f * (c0[r] + c1[r]);      // x^2 term constant per row -> dropped
            if (v < runmin[r]) { runmin[r] = v; runidx[r] = code; }
        }
    }

    // ---- cross-lane argmin within each 16-lane half (xor 1,2,4,8 stays in half) ----
#pragma unroll
    for (int r = 0; r < 8; ++r) {
        float mv = runmin[r];
        int   iv = runidx[r];
#pragma unroll
        for (int off = 1; off < 16; off <<= 1) {
            float om = __shfl_xor(mv, off, 32);
            int   oi = __shfl_xor(iv, off, 32);
            if (om < mv || (om == mv && oi < iv)) { mv = om; iv = oi; }
        }
        if (mrow == 0) {                            // row M = r + 8*half
            wmin[wave][r + 8 * half] = mv;
            widx[wave][r + 8 * half] = iv;
        }
    }
    __syncthreads();

    // ---- reduce across the 8 waves (waves ordered by increasing code -> first-min wins) ----
    if (t < 16) {
        float mv = wmin[0][t];
        int   iv = widx[0][t];
#pragma unroll
        for (int w = 1; w < 8; ++w) {
            float om = wmin[w][t];
            if (om < mv) { mv = om; iv = widx[w][t]; }
        }
        fidx[t] = iv;
    }
    __syncthreads();

    // ---- cw gather + update_dict scatter (thread: 4 consecutive e of one row) ----
    {
        int m = t >> 4;
        int e = (t & 15) * 4;
        int k = fidx[m];
        f4 v = *(const f4*)(cbd + (size_t)k * EMB + e);
        __builtin_nontemporal_store(v, (f4*)(cw + (size_t)(b0 + m) * (DIMC * EMB) + d * EMB + e));
        float* urow = upd + ((size_t)d * BOOK + k) * EMB + e;
        const float* lr = &lx[m * 66 + e];
        atomicAdd(urow + 0, lr[0]);
        atomicAdd(urow + 1, lr[1]);
        atomicAdd(urow + 2, lr[2]);
        atomicAdd(urow + 3, lr[3]);
    }
    if (t < 16) atomicAdd(&counts[d * BOOK + fidx[t]], 1.0f);

    // ---- one_hot rows: 16 rows x 1024 zeros + single 1.0, NT B128 streamout ----
    {
        int m  = t >> 4;
        int c0i = (t & 15) * 64;
        int k  = fidx[m];
        float* dst = onehot + ((size_t)(b0 + m) * DIMC + d) * BOOK + c0i;
#pragma unroll 4
        for (int c = 0; c < 64; c += 4) {
            f4 z = {0.f, 0.f, 0.f, 0.f};
            int cc = c0i + c;
            if (k >= cc && k < cc + 4) {
                if      (k == cc)     z.x = 1.0f;
                else if (k == cc + 1) z.y = 1.0f;
                else if (k == cc + 2) z.z = 1.0f;
                else                  z.w = 1.0f;
            }
            __builtin_nontemporal_store(z, (f4*)(dst + c));
        }
    }
}

// ---------------- EMA finalize: new_codebook + new_ema_counts ----------------
__global__ void vq_finalize_kernel(const float* __restrict__ cb,
                                   const float* __restrict__ ema,
                                   const float* __restrict__ counts,
                                   const float* __restrict__ upd,
                                   float* __restrict__ ncb,
                                   float* __restrict__ nema) {
    int dk = blockIdx.x * 256 + threadIdx.x;          // 65536 (d,k) rows
    float e0 = ema[dk];
    nema[dk] = DECAYF * e0 + GAINF * counts[dk];
    float inv = GAINF / (e0 + EPSF);
    const f4* c4 = (const f4*)(cb  + (size_t)dk * EMB);
    const f4* u4 = (const f4*)(upd + (size_t)dk * EMB);
    f4*       o4 = (f4*)(ncb + (size_t)dk * EMB);
#pragma unroll
    for (int j = 0; j < 16; ++j) {
        f4 c = c4[j], u = u4[j], r;
        r.x = c.x * DECAYF + u.x * inv;
        r.y = c.y * DECAYF + u.y * inv;
        r.z = c.z * DECAYF + u.z * inv;
        r.w = c.w * DECAYF + u.w * inv;
        __builtin_nontemporal_store(r, o4 + j);
    }
}

extern "C" void kernel_launch(void* const* d_in, const int* in_sizes, int n_in,
                              void* d_out, int out_size, void* d_ws, size_t ws_size,
                              hipStream_t stream) {
    const float* x   = (const float*)d_in[0];
    const float* cb  = (const float*)d_in[1];
    const float* ema = (const float*)d_in[2];

    float* out    = (float*)d_out;
    float* cw     = out;                                        // [512,4096]
    float* onehot = cw  + (size_t)BATCH * DIMC * EMB;           // [512,64,1024]
    float* ncb    = onehot + (size_t)BATCH * DIMC * BOOK;       // [64,1024,64]
    float* nema   = ncb + (size_t)DIMC * BOOK * EMB;            // [64,1024]

    float* ws     = (float*)d_ws;
    float* bsq    = ws;                                         // 65536
    float* counts = bsq + DIMC * BOOK;                          // 65536
    float* upd    = counts + DIMC * BOOK;                       // 4194304

    int nz = DIMC * BOOK + DIMC * BOOK * EMB;                   // counts + upd
    vq_zero_kernel<<<(nz / 4 + 255) / 256, 256, 0, stream>>>(counts, nz);
    vq_bsq_kernel<<<(DIMC * BOOK) / 256, 256, 0, stream>>>(cb, bsq);

    dim3 grid(BATCH / 16, DIMC);
    vq_main_kernel<<<grid, 256, 0, stream>>>(x, cb, bsq, cw, onehot, counts, upd);

    vq_finalize_kernel<<<(DIMC * BOOK) / 256, 256, 0, stream>>>(cb, ema, counts, upd, ncb, nema);
}